// ProtoNetLoss_40647570489383
// MI455X (gfx1250) — compile-verified
//
#include <hip/hip_runtime.h>
#include <hip/hip_bf16.h>
#include <math.h>

// Problem constants (from reference)
#define NQc 2048
#define NSc 65536
#define Kc  512
#define Dc  128

typedef __attribute__((ext_vector_type(2))) float v2f;
typedef __attribute__((ext_vector_type(8))) float v8f;

// ---------------------------------------------------------------------------
// Kernel 1: zero workspace accumulators and output scalar
// ---------------------------------------------------------------------------
__global__ void pn_zero_kernel(float* __restrict__ mus, float* __restrict__ cnt,
                               float* __restrict__ out) {
    int i = blockIdx.x * blockDim.x + threadIdx.x;
    if (i < Kc * Dc) mus[i] = 0.0f;
    if (i < Kc)      cnt[i] = 0.0f;
    if (i == 0)      out[0] = 0.0f;
}

// ---------------------------------------------------------------------------
// Kernel 2: class sums (mus = one_hot(ys)^T @ xs) + class counts via atomics.
// 32 threads per support point; each thread handles 4 contiguous dims.
// ---------------------------------------------------------------------------
__global__ void pn_scatter_kernel(const float* __restrict__ xs,
                                  const int* __restrict__ ys,
                                  float* __restrict__ mus,
                                  float* __restrict__ cnt) {
    int gtid = blockIdx.x * blockDim.x + threadIdx.x;
    int s = gtid >> 5;            // support index
    int j = gtid & 31;            // dim group
    if (s >= NSc) return;
    int c = ys[s];
    int d = j * 4;
    const float4 v = *(const float4*)(xs + (size_t)s * Dc + d);
    float* dst = mus + (size_t)c * Dc + d;
    atomicAdd(dst + 0, v.x);
    atomicAdd(dst + 1, v.y);
    atomicAdd(dst + 2, v.z);
    atomicAdd(dst + 3, v.w);
    if (j == 0) atomicAdd(cnt + c, 1.0f);
}

// ---------------------------------------------------------------------------
// Kernel 3: P[k] = mus[k]/max(cnt[k],0.1), stored transposed PT[d][k] for
// coalesced WMMA B-fragment loads; pn[k] = ||P[k]||^2.
// One block per class, 128 threads (one per dim).
// ---------------------------------------------------------------------------
__global__ void pn_proto_kernel(const float* __restrict__ mus,
                                const float* __restrict__ cnt,
                                float* __restrict__ PT,
                                float* __restrict__ pn) {
    int k = blockIdx.x;
    int d = threadIdx.x;
    float c = cnt[k];
    float p = mus[(size_t)k * Dc + d] / fmaxf(c, 0.1f);
    PT[(size_t)d * Kc + k] = p;

    // reduce p*p over 128 threads (4 waves)
    float sq = p * p;
    for (int off = 16; off > 0; off >>= 1)
        sq += __shfl_xor(sq, off, 32);
    __shared__ float red[4];
    int wave = d >> 5;
    if ((d & 31) == 0) red[wave] = sq;
    __syncthreads();
    if (d == 0) pn[k] = red[0] + red[1] + red[2] + red[3];
}

// ---------------------------------------------------------------------------
// Kernel 4: main fused kernel.
// Block = 16 queries x all 512 classes. 256 threads = 8 waves.
// Wave w computes a 16x64 logit strip with V_WMMA_F32_16X16X4_F32
// (32 chained WMMAs over D=128 per 16x16 tile), applies the distance
// transform + (cnt>0.1) mask, then per-query leave-one-out correction,
// then a shuffle-reduced softmax and nll accumulation.
// ---------------------------------------------------------------------------
__global__ void __launch_bounds__(256)
pn_main_kernel(const float* __restrict__ xq,
               const int* __restrict__ ys,
               const int* __restrict__ pos,
               const float* __restrict__ mus,
               const float* __restrict__ cnt,
               const float* __restrict__ PT,
               const float* __restrict__ pn,
               float* __restrict__ out) {
    __shared__ float s_xq[16][132];      // padded: bank-conflict-free A reads
    __shared__ float s_logit[16][Kc];    // 16 x 512 logit strip
    __shared__ float s_qn[16];

    const int tid  = threadIdx.x;
    const int wave = tid >> 5;
    const int lane = tid & 31;
    const int m    = lane & 15;          // row (A) / col (B,D) within tile
    const int hi   = lane >> 4;          // lane-half: selects K pair / D rows
    const int q0   = blockIdx.x * 16;    // first query of this block

    // ---- stage xq tile into LDS (coalesced) ----
    for (int idx = tid; idx < 16 * Dc; idx += 256) {
        int r = idx >> 7;                // /128
        int cdim = idx & (Dc - 1);
        s_xq[r][cdim] = xq[(size_t)(q0 + r) * Dc + cdim];
    }
    __syncthreads();

    // ---- query norms ----
    if (tid < 16) {
        float acc = 0.0f;
        #pragma unroll 8
        for (int d = 0; d < Dc; ++d) acc += s_xq[tid][d] * s_xq[tid][d];
        s_qn[tid] = acc;
    }
    __syncthreads();

    // ---- GEMM: G = xq_tile @ P^T via WMMA f32 16x16x4 ----
    // wave w owns class columns [w*64, w*64+64): 4 tiles of 16 columns.
    for (int ct = 0; ct < 4; ++ct) {
        const int col = wave * 64 + ct * 16 + m;   // global class column
        v8f acc = {};
        #pragma unroll 4
        for (int ks = 0; ks < 32; ++ks) {
            const int kk = ks * 4 + 2 * hi;
            v2f a, b;
            a.x = s_xq[m][kk];
            a.y = s_xq[m][kk + 1];
            b.x = PT[(size_t)kk * Kc + col];
            b.y = PT[(size_t)(kk + 1) * Kc + col];
            acc = __builtin_amdgcn_wmma_f32_16x16x4_f32(
                false, a, false, b, (short)0, acc, false, false);
        }
        // transform: L = -0.5*(qn - 2G + pn) * (cnt > 0.1)
        const float pnv  = pn[col];
        const float mask = (cnt[col] > 0.1f) ? 1.0f : 0.0f;
        #pragma unroll
        for (int v = 0; v < 8; ++v) {
            const int row = v + 8 * hi;
            const float g = acc[v];
            s_logit[row][col] = mask * (-0.5f * (s_qn[row] - 2.0f * g + pnv));
        }
    }
    __syncthreads();

    // ---- leave-one-out correction: wave w fixes queries 2w, 2w+1 ----
    for (int qq = wave * 2; qq < wave * 2 + 2; ++qq) {
        const int q  = q0 + qq;
        const int c  = ys[pos[q]];
        const float Cc    = cnt[c] - 1.0f;
        const float denom = fmaxf(Cc, 0.1f);
        const int d = lane * 4;
        const float4 mu = *(const float4*)(mus + (size_t)c * Dc + d);
        const float x0 = s_xq[qq][d + 0], x1 = s_xq[qq][d + 1];
        const float x2 = s_xq[qq][d + 2], x3 = s_xq[qq][d + 3];
        float df0 = x0 - (mu.x - x0) / denom;
        float df1 = x1 - (mu.y - x1) / denom;
        float df2 = x2 - (mu.z - x2) / denom;
        float df3 = x3 - (mu.w - x3) / denom;
        float part = df0 * df0 + df1 * df1 + df2 * df2 + df3 * df3;
        for (int off = 16; off > 0; off >>= 1)
            part += __shfl_xor(part, off, 32);
        if (lane == 0) {
            float l = -0.5f * part;
            if (!(Cc > 0.1f)) l = 0.0f;
            s_logit[qq][c] = l;
        }
    }
    __syncthreads();

    // ---- softmax + nll (T = 1): wave w handles queries 2w, 2w+1 ----
    for (int qq = wave * 2; qq < wave * 2 + 2; ++qq) {
        float mx = -1e30f;
        for (int k = lane; k < Kc; k += 32) mx = fmaxf(mx, s_logit[qq][k]);
        for (int off = 16; off > 0; off >>= 1)
            mx = fmaxf(mx, __shfl_xor(mx, off, 32));
        float sum = 0.0f;
        for (int k = lane; k < Kc; k += 32) sum += __expf(s_logit[qq][k] - mx);
        for (int off = 16; off > 0; off >>= 1)
            sum += __shfl_xor(sum, off, 32);
        if (lane == 0) {
            const int c = ys[pos[q0 + qq]];
            const float nll = -(s_logit[qq][c] - mx - __logf(sum));
            atomicAdd(out, nll * (1.0f / (float)NQc));
        }
    }
}

// ---------------------------------------------------------------------------
// Launch
// ---------------------------------------------------------------------------
extern "C" void kernel_launch(void* const* d_in, const int* in_sizes, int n_in,
                              void* d_out, int out_size, void* d_ws, size_t ws_size,
                              hipStream_t stream) {
    const float* xq  = (const float*)d_in[0];
    // d_in[1] = yq (unused by the reference computation)
    const float* xs  = (const float*)d_in[2];
    const int*   ys  = (const int*)d_in[3];
    const int*   pos = (const int*)d_in[4];
    float* out = (float*)d_out;

    // workspace layout (floats): mus[K*D] | cnt[K] | pn[K] | PT[D*K]
    float* mus = (float*)d_ws;
    float* cnt = mus + (size_t)Kc * Dc;
    float* pn  = cnt + Kc;
    float* PT  = pn + Kc;

    // 1) zero accumulators + output
    pn_zero_kernel<<<(Kc * Dc + 255) / 256, 256, 0, stream>>>(mus, cnt, out);

    // 2) scatter class sums: NS points x 32 threads
    pn_scatter_kernel<<<(NSc * 32) / 256, 256, 0, stream>>>(xs, ys, mus, cnt);

    // 3) prototypes (transposed) + norms
    pn_proto_kernel<<<Kc, Dc, 0, stream>>>(mus, cnt, PT, pn);

    // 4) fused WMMA distance-logits + LOO correction + softmax + mean-nll
    pn_main_kernel<<<NQc / 16, 256, 0, stream>>>(xq, ys, pos, mus, cnt, PT, pn, out);
}